// ScatterEdges_7971459302125
// MI455X (gfx1250) — compile-verified
//
#include <hip/hip_runtime.h>

// ScatterEdges for MI455X (gfx1250, wave32).
//
// Strategy (memory/atomic-bound op; WMMA not applicable — no matmul shape):
//  * x [E,48] f32 is streamed exactly once: nontemporal float4 loads
//    (global_load_b128 th:NT) so the 307 MB stream does not pollute L2,
//    keeping the 19.2 MB output resident in the 192 MB L2 for atomics.
//  * 12 lanes per edge, one float4 chunk each -> fully coalesced 512B per wave.
//  * 2 x 4 non-returning global_atomic_add_f32 per thread (STOREcnt path,
//    no data return to wait on).
//  * Kernel 0 zeroes d_out (harness poisons it with 0xAA).

typedef float v4f __attribute__((ext_vector_type(4)));

__global__ void __launch_bounds__(256)
ZeroOut_kernel(v4f* __restrict__ out4, int n4) {
    int i = blockIdx.x * 256 + threadIdx.x;
    if (i < n4) {
        v4f z = {0.0f, 0.0f, 0.0f, 0.0f};
        out4[i] = z;
    }
}

__global__ void __launch_bounds__(256)
ScatterEdges_7971459302125_kernel(const v4f* __restrict__ x4,
                                  const float* __restrict__ sw,
                                  const long long* __restrict__ esrc,
                                  const long long* __restrict__ edst,
                                  float* __restrict__ out,
                                  int n_edges) {
    long long t = (long long)blockIdx.x * 256 + threadIdx.x;
    int e = (int)(t / 12);        // edge index
    int c = (int)(t - (long long)e * 12);  // float4 chunk within the 48-float row
    if (e >= n_edges) return;

    // Streamed once: non-temporal 128-bit load (th:NT) -> don't cache in L2.
    v4f v = __builtin_nontemporal_load(x4 + (long long)e * 12 + c);

    // Per-edge scalars: regular loads, broadcast-shared by the 12 lanes of
    // this edge through L0/L2.
    float s = sw[e];
    v *= s;

    long long si = esrc[e];
    long long di = edst[e];

    float* ps = out + si * 48 + (c << 2);
    float* pd = out + di * 48 + (c << 2);

    // Non-returning relaxed agent-scope f32 atomics -> global_atomic_add_f32
    // (no return VGPR, tracked on STOREcnt). Output rows live in L2.
    __hip_atomic_fetch_add(ps + 0, v.x, __ATOMIC_RELAXED, __HIP_MEMORY_SCOPE_AGENT);
    __hip_atomic_fetch_add(ps + 1, v.y, __ATOMIC_RELAXED, __HIP_MEMORY_SCOPE_AGENT);
    __hip_atomic_fetch_add(ps + 2, v.z, __ATOMIC_RELAXED, __HIP_MEMORY_SCOPE_AGENT);
    __hip_atomic_fetch_add(ps + 3, v.w, __ATOMIC_RELAXED, __HIP_MEMORY_SCOPE_AGENT);
    __hip_atomic_fetch_add(pd + 0, v.x, __ATOMIC_RELAXED, __HIP_MEMORY_SCOPE_AGENT);
    __hip_atomic_fetch_add(pd + 1, v.y, __ATOMIC_RELAXED, __HIP_MEMORY_SCOPE_AGENT);
    __hip_atomic_fetch_add(pd + 2, v.z, __ATOMIC_RELAXED, __HIP_MEMORY_SCOPE_AGENT);
    __hip_atomic_fetch_add(pd + 3, v.w, __ATOMIC_RELAXED, __HIP_MEMORY_SCOPE_AGENT);
}

extern "C" void kernel_launch(void* const* d_in, const int* in_sizes, int n_in,
                              void* d_out, int out_size, void* d_ws, size_t ws_size,
                              hipStream_t stream) {
    // setup_inputs() order: x [E*48] f32, switch [E] f32, species [N] i32,
    // edge_src [E] i64, edge_dst [E] i64.
    const v4f*       x4   = (const v4f*)d_in[0];
    const float*     sw   = (const float*)d_in[1];
    const long long* esrc = (const long long*)d_in[3];
    const long long* edst = (const long long*)d_in[4];
    float*           out  = (float*)d_out;

    const int E = in_sizes[1];          // 1,600,000
    // out_size = N * 48, divisible by 4.
    const int n4 = out_size / 4;

    // 1) Zero the (poisoned) output.
    int zblocks = (n4 + 255) / 256;
    ZeroOut_kernel<<<zblocks, 256, 0, stream>>>((v4f*)out, n4);

    // 2) Scatter-add: 12 threads per edge (one float4 chunk each).
    long long total = (long long)E * 12;
    int sblocks = (int)((total + 255) / 256);
    ScatterEdges_7971459302125_kernel<<<sblocks, 256, 0, stream>>>(
        x4, sw, esrc, edst, out, E);
}